// NeuralSplineTransformer_77704548319330
// MI455X (gfx1250) — compile-verified
//
#include <hip/hip_runtime.h>
#include <hip/hip_bf16.h>
#include <math.h>

#define B_ 2048
#define F_ 512
#define K_ 16
#define P_ 49              // 3*K + 1
#define THREADS 512        // 16 wave32 per block, one block per batch row

__global__ __launch_bounds__(THREADS)
void nsf_kernel(const float* __restrict__ x,
                const float* __restrict__ params,   // (B, P, F)
                const float* __restrict__ x0v,      // (F)
                const float* __restrict__ xfv,      // (F)
                float* __restrict__ y_out,          // (B, F)
                float* __restrict__ ld_out)         // (B)
{
    __shared__ __align__(16) float sp[P_ * F_];     // 100,352 B staged params
    __shared__ float partials[THREADS / 32];

    const int b   = blockIdx.x;
    const int tid = threadIdx.x;
    const int f   = tid;

    // ---------------- async stage: params[b,:,:] -> LDS (non-temporal) -------
    // 205 MB streamed once > 192 MB L2: NT keeps it from evicting reusable data.
    const float* gbase = params + (size_t)b * (P_ * F_);
    const uint32_t lds_base = (uint32_t)(uintptr_t)(&sp[0]);  // flat low 32b == LDS offset
    const int n16 = (P_ * F_ * 4) / 16;                       // 6272 x 16B packets
    for (int i = tid; i < n16; i += THREADS) {
        uint32_t off = (uint32_t)i * 16u;
        uint32_t dst = lds_base + off;
        asm volatile("global_load_async_to_lds_b128 %0, %1, %2 th:TH_LOAD_NT"
                     :: "v"(dst), "v"(off), "s"(gbase)
                     : "memory");
    }

    // Overlap the small per-thread loads with the async DMA: issue them BEFORE
    // the asynccnt wait ("memory" clobber pins ordering otherwise).
    const float xb  = __builtin_nontemporal_load(&x[(size_t)b * F_ + f]); // read-once
    const float x0f = x0v[f];                                             // reused by all blocks
    const float xff = xfv[f];
    const float range = xff - x0f;

    asm volatile("s_wait_asynccnt 0" ::: "memory");
    __syncthreads();

    // ---------------- per-element spline math ----------------
    float ew[K_], eh[K_], sl[K_ + 1];

    // softmax (max-subtracted) for widths & heights
    float mw = -__builtin_inff(), mh = -__builtin_inff();
#pragma unroll
    for (int i = 0; i < K_; ++i) {
        ew[i] = sp[i * F_ + f];
        eh[i] = sp[(K_ + i) * F_ + f];
        mw = fmaxf(mw, ew[i]);
        mh = fmaxf(mh, eh[i]);
    }
    float sw = 0.f, sh = 0.f;
#pragma unroll
    for (int i = 0; i < K_; ++i) {
        ew[i] = __expf(ew[i] - mw); sw += ew[i];
        eh[i] = __expf(eh[i] - mh); sh += eh[i];
    }
    const float scw = range / sw;
    const float sch = range / sh;
#pragma unroll
    for (int i = 0; i < K_; ++i) { ew[i] *= scw; eh[i] *= sch; }  // widths, heights

    // softplus slopes (stable form)
#pragma unroll
    for (int i = 0; i <= K_; ++i) {
        float z = sp[(2 * K_ + i) * F_ + f];
        sl[i] = fmaxf(z, 0.f) + log1pf(__expf(-fabsf(z)));
    }

    // O(K) comparison-count bin search: bin = clip(sum(x > knots_x) - 1, 0, K-1)
    int cnt = 1;                   // x > x0 always (x strictly inside)
    float cum = 0.f;
#pragma unroll
    for (int i = 0; i < K_; ++i) {
        cum += ew[i];
        cnt += (xb > x0f + cum) ? 1 : 0;
    }
    const int bin = min(cnt - 1, K_ - 1);

    // predicated gather of prefix sums and bin values (no dynamic reg indexing)
    float px = 0.f, py = 0.f, wb = 0.f, hb = 0.f;
#pragma unroll
    for (int i = 0; i < K_; ++i) {
        px += (i < bin) ? ew[i] : 0.f;
        py += (i < bin) ? eh[i] : 0.f;
        wb  = (i == bin) ? ew[i] : wb;
        hb  = (i == bin) ? eh[i] : hb;
    }
    float d0 = 0.f, d1 = 0.f;
#pragma unroll
    for (int i = 0; i <= K_; ++i) {
        d0 = (i == bin)     ? sl[i] : d0;
        d1 = (i == bin + 1) ? sl[i] : d1;
    }

    const float xk  = x0f + px;
    const float yk  = x0f + py;             // y0 == x0 (module default)
    const float s   = hb / wb;
    const float eps = (xb - xk) / wb;
    const float om  = 1.f - eps;
    const float e1e = eps * om;
    const float e2  = eps * eps;
    const float den = s + (d1 + d0 - 2.f * s) * e1e;
    const float y   = yk + hb * (s * e2 + d0 * e1e) / den;
    const float dy  = s * s * (d1 * e2 + 2.f * s * e1e + d0 * om * om) / (den * den);

    __builtin_nontemporal_store(y, &y_out[(size_t)b * F_ + f]);   // write-once stream

    // ---------------- deterministic block reduction of log(dy) ----------------
    float lg = __logf(dy);
#pragma unroll
    for (int off = 16; off >= 1; off >>= 1)
        lg += __shfl_down(lg, off, 32);
    if ((tid & 31) == 0) partials[tid >> 5] = lg;
    __syncthreads();
    if (tid < 32) {                          // whole wave 0 participates in shuffles
        float v = (tid < (THREADS / 32)) ? partials[tid] : 0.f;
#pragma unroll
        for (int off = 8; off >= 1; off >>= 1)
            v += __shfl_down(v, off, 32);
        if (tid == 0) ld_out[b] = v;
    }
}

extern "C" void kernel_launch(void* const* d_in, const int* in_sizes, int n_in,
                              void* d_out, int out_size, void* d_ws, size_t ws_size,
                              hipStream_t stream) {
    const float* x      = (const float*)d_in[0];
    const float* params = (const float*)d_in[1];
    const float* x0     = (const float*)d_in[2];
    const float* xf     = (const float*)d_in[3];
    float* y  = (float*)d_out;                    // (B,F) flat
    float* ld = y + (size_t)B_ * F_;              // then (B,)
    nsf_kernel<<<dim3(B_), dim3(THREADS), 0, stream>>>(x, params, x0, xf, y, ld);
}